// LightGCN_79216376807726
// MI455X (gfx1250) — compile-verified
//
#include <hip/hip_runtime.h>
#include <cstdint>
#include <cstddef>

#define D_DIM 64
#define EPS_F 1e-12f

typedef float v2f __attribute__((ext_vector_type(2)));
typedef float v8f __attribute__((ext_vector_type(8)));

// ---------------------------------------------------------------------------
// Init: cur = emb, res = emb, agg = 0  (one side at a time)
// ---------------------------------------------------------------------------
__global__ void lgcn_init_kernel(const float* __restrict__ emb,
                                 float* __restrict__ cur,
                                 float* __restrict__ agg,
                                 float* __restrict__ res,
                                 int n) {
    int i = blockIdx.x * blockDim.x + threadIdx.x;
    if (i < n) {
        float v = emb[i];
        cur[i] = v;
        res[i] = v;
        agg[i] = 0.0f;
    }
}

// ---------------------------------------------------------------------------
// Edge scatter: one wave32 per edge. lane L handles dims {2L, 2L+1}.
//   agg_e[e] += w * cur_u[u]   and   agg_u[u] += w * cur_e[e]
// L2-resident gathers + native global_atomic_add_f32 scatters.
// ---------------------------------------------------------------------------
__global__ void lgcn_scatter_kernel(const float* __restrict__ cur_u,
                                    const float* __restrict__ cur_e,
                                    const int*   __restrict__ eu,
                                    const int*   __restrict__ ee,
                                    const float* __restrict__ ew,
                                    float* __restrict__ agg_u,
                                    float* __restrict__ agg_e,
                                    int nnz) {
    int wid  = (blockIdx.x * blockDim.x + threadIdx.x) >> 5;
    int lane = threadIdx.x & 31;
    if (wid >= nnz) return;                       // whole-wave uniform exit

    int   u = eu[wid];
    int   e = ee[wid];
    float w = ew[wid];

    const float2* ru = reinterpret_cast<const float2*>(cur_u + (size_t)u * D_DIM);
    const float2* re = reinterpret_cast<const float2*>(cur_e + (size_t)e * D_DIM);
    float2 a = ru[lane];
    float2 b = re[lane];

    float* oe = agg_e + (size_t)e * D_DIM + 2 * lane;
    float* ou = agg_u + (size_t)u * D_DIM + 2 * lane;
    unsafeAtomicAdd(oe,     w * a.x);
    unsafeAtomicAdd(oe + 1, w * a.y);
    unsafeAtomicAdd(ou,     w * b.x);
    unsafeAtomicAdd(ou + 1, w * b.y);
}

// ---------------------------------------------------------------------------
// Normalize 16 rows per wave via v_wmma_f32_16x16x4_f32 Gram trick:
//   f32 A(16x4) layout: VGPR0 = K{0,2}, VGPR1 = K{1,3}, M = lane%16
//   f32 B(4x16) layout: VGPR0 = K{0,2}, VGPR1 = K{1,3}, N = lane%16
//   => same registers serve as A (=X chunk) and B (=X^T chunk),
//      accumulating D = X·X^T over 16 K-chunks; diag(D) = row sumsq.
// Then: cur = agg * 1/max(sqrt(ss), eps);  res += cur;  agg = 0 (next hop).
// ---------------------------------------------------------------------------
__global__ void lgcn_normalize_kernel(float* __restrict__ agg,   // read + zeroed
                                      float* __restrict__ cur,
                                      float* __restrict__ res,
                                      int nrows) {
    __shared__ float gram[8][16 * 16];            // per-wave 16x16 Gram tile

    int wid  = threadIdx.x >> 5;
    int lane = threadIdx.x & 31;
    int tile = blockIdx.x * (blockDim.x >> 5) + wid;
    if (tile * 16 >= nrows) return;               // whole-wave uniform exit

    int row0   = tile * 16;
    int r      = lane & 15;                       // row within tile
    int koff   = (lane < 16) ? 0 : 2;             // K sub-offset per A/B layout
    int rowClm = (row0 + r < nrows) ? (row0 + r) : (nrows - 1); // safe tail read
    bool valid = (row0 + r) < nrows;

    const float* src = agg + (size_t)rowClm * D_DIM + koff;

    v8f c = {0.f, 0.f, 0.f, 0.f, 0.f, 0.f, 0.f, 0.f};
    float2 vals[16];
#pragma unroll
    for (int i = 0; i < 16; ++i) {
        float2 t = *reinterpret_cast<const float2*>(src + 4 * i);
        vals[i] = t;
        v2f a; a.x = t.x; a.y = t.y;
        // (neg_a, A, neg_b, B, c_mod, C, reuse_a, reuse_b); A==B => X·X^T
        c = __builtin_amdgcn_wmma_f32_16x16x4_f32(false, a, false, a,
                                                  (short)0, c, false, false);
    }

    // Spill Gram tile to this wave's LDS region; wave-ordered LDS makes the
    // subsequent diagonal load see the stores without a block barrier.
    float* g  = gram[wid];
    int    mb = (lane < 16) ? 0 : 8;              // C/D layout: M = r (+8 hi half)
    int    nn = lane & 15;                        // N = lane%16
#pragma unroll
    for (int rr = 0; rr < 8; ++rr)
        g[(mb + rr) * 16 + nn] = c[rr];
    __builtin_amdgcn_wave_barrier();              // scheduling fence only

    float ss    = g[r * 16 + r];                  // ||row r||^2
    float scale = 1.0f / fmaxf(sqrtf(ss), EPS_F);

    float* curp = cur + (size_t)rowClm * D_DIM + koff;
    float* resp = res + (size_t)rowClm * D_DIM + koff;
    float* aggz = agg + (size_t)rowClm * D_DIM + koff;
    if (valid) {
#pragma unroll
        for (int i = 0; i < 16; ++i) {
            float2 o;
            o.x = vals[i].x * scale;
            o.y = vals[i].y * scale;
            *reinterpret_cast<float2*>(curp + 4 * i) = o;
            float2 rv = *reinterpret_cast<float2*>(resp + 4 * i);
            rv.x += o.x;
            rv.y += o.y;
            *reinterpret_cast<float2*>(resp + 4 * i) = rv;
            float2 z; z.x = 0.f; z.y = 0.f;       // re-zero agg for next hop
            *reinterpret_cast<float2*>(aggz + 4 * i) = z;
        }
    }
}

// ---------------------------------------------------------------------------
// Host launcher
// ---------------------------------------------------------------------------
extern "C" void kernel_launch(void* const* d_in, const int* in_sizes, int n_in,
                              void* d_out, int out_size, void* d_ws, size_t ws_size,
                              hipStream_t stream) {
    const float* user_emb   = (const float*)d_in[0];
    const float* entity_emb = (const float*)d_in[1];
    const int*   edge_u     = (const int*)d_in[2];
    const int*   edge_e     = (const int*)d_in[3];
    const float* edge_w     = (const float*)d_in[4];

    const int n_user_el = in_sizes[0];            // N_USERS * 64
    const int n_ent_el  = in_sizes[1];            // N_ENTITIES * 64
    const int nnz       = in_sizes[2];
    const int n_users   = n_user_el / D_DIM;
    const int n_ents    = n_ent_el / D_DIM;

    // Output: (entity_res, user_res) concatenated flat in return order.
    float* res_e = (float*)d_out;
    float* res_u = res_e + (size_t)n_ent_el;

    // Workspace: cur_u | cur_e | agg_u | agg_e  (4 x 25.6 MB)
    float* ws    = (float*)d_ws;
    float* cur_u = ws;
    float* cur_e = cur_u + (size_t)n_user_el;
    float* agg_u = cur_e + (size_t)n_ent_el;
    float* agg_e = agg_u + (size_t)n_user_el;

    const int TB = 256;

    // Init both sides: cur = emb, res = emb, agg = 0
    lgcn_init_kernel<<<(n_user_el + TB - 1) / TB, TB, 0, stream>>>(
        user_emb, cur_u, agg_u, res_u, n_user_el);
    lgcn_init_kernel<<<(n_ent_el + TB - 1) / TB, TB, 0, stream>>>(
        entity_emb, cur_e, agg_e, res_e, n_ent_el);

    const int wavesPerBlock = TB / 32;
    const int scatterBlocks = (nnz + wavesPerBlock - 1) / wavesPerBlock;
    const int tilesE = (n_ents  + 15) / 16;
    const int tilesU = (n_users + 15) / 16;
    const int normBlocksE = (tilesE + wavesPerBlock - 1) / wavesPerBlock;
    const int normBlocksU = (tilesU + wavesPerBlock - 1) / wavesPerBlock;

    for (int hop = 0; hop < 3; ++hop) {
        lgcn_scatter_kernel<<<scatterBlocks, TB, 0, stream>>>(
            cur_u, cur_e, edge_u, edge_e, edge_w, agg_u, agg_e, nnz);
        lgcn_normalize_kernel<<<normBlocksE, TB, 0, stream>>>(
            agg_e, cur_e, res_e, n_ents);
        lgcn_normalize_kernel<<<normBlocksU, TB, 0, stream>>>(
            agg_u, cur_u, res_u, n_users);
    }
}